// GSeparableKernel_65884798320718
// MI455X (gfx1250) — compile-verified
//
#include <hip/hip_runtime.h>
#include <math.h>

typedef __attribute__((ext_vector_type(2))) float v2f;
typedef __attribute__((ext_vector_type(8))) float v8f;

#define O_DIM 512
#define CIN_DIM 512
#define K_H 16
#define NI_DIM 16
#define NO_DIM 16
#define KS_DIM 7

__device__ __forceinline__ float selw(int k, int i0, int i1, float f) {
  float w = (k == i0) ? (1.0f - f) : 0.0f;
  return (k == i1) ? f : w;
}

// ---------------------------------------------------------------------------
// Output 1: weight_H_out[o, ni, cin, no] as a M=262144 x K=16 x N=256 GEMM
// against the circular-lerp selection matrix, via chained f32 WMMAs.
// Each wave owns one 16-row (cin) tile, loads A once, loops all 16 ni tiles.
// The 256 MB output stream is stored with the NT temporal hint so it does not
// evict the 16 MB weight_H working set (reused 16x per tile) from the 192 MB L2.
// ---------------------------------------------------------------------------
__global__ void __launch_bounds__(256)
gsep_lerp_wmma_kernel(const float* __restrict__ wH,   // [O, CIN, K]
                      const float* __restrict__ inH,  // [NI]
                      const float* __restrict__ outH, // [NO]
                      float* __restrict__ out1)       // [O, NI, CIN, NO]
{
  const int lane = threadIdx.x & 31;
  const int wave = threadIdx.x >> 5;
  const int tile = blockIdx.x * 8 + wave;   // tile over flattened (o,cin) rows
  const int row0 = tile << 4;               // first (o,cin) row of this tile
  const int o    = row0 >> 9;               // row0 / 512  (tiles never cross o)
  const int cin0 = row0 & 511;

  const int h = lane >> 4;                  // half-wave: k-subcolumn select
  const int n = lane & 15;                  // A row id / B,D column id

  // A tile: row M=n of weight_H, split across 4 K=4 WMMAs.
  // WMMA j, VGPR v, half h  ->  k = 4j + 2h + v  => one v2f per j.
  const float* rp = wH + ((size_t)(row0 + n)) * K_H + (h << 1);
  v2f a0 = *(const v2f*)(rp + 0);
  v2f a1 = *(const v2f*)(rp + 4);
  v2f a2 = *(const v2f*)(rp + 8);
  v2f a3 = *(const v2f*)(rp + 12);

  const float oang = outH[n];
  const float TWO_PI = 6.28318530717958647692f;

  for (int ni = 0; ni < NI_DIM; ++ni) {
    // circular lerp coefficients for the (ni, no=n) pair
    float d   = inH[ni] - oang;
    float t   = d - TWO_PI * floorf(d * (1.0f / TWO_PI));   // mod 2pi
    float u   = t * ((float)K_H / TWO_PI);                   // / step
    float i0f = floorf(u);
    float f   = u - i0f;
    int   i0  = ((int)i0f) & (K_H - 1);
    int   i1  = (i0 + 1) & (K_H - 1);

    // B tile (selection matrix column no=n), same k indexing as A
    const int kb = h << 1;
    v2f b0, b1, b2, b3;
    b0.x = selw(kb + 0,  i0, i1, f);  b0.y = selw(kb + 1,  i0, i1, f);
    b1.x = selw(kb + 4,  i0, i1, f);  b1.y = selw(kb + 5,  i0, i1, f);
    b2.x = selw(kb + 8,  i0, i1, f);  b2.y = selw(kb + 9,  i0, i1, f);
    b3.x = selw(kb + 12, i0, i1, f);  b3.y = selw(kb + 13, i0, i1, f);

    v8f c = {};
    c = __builtin_amdgcn_wmma_f32_16x16x4_f32(false, a0, false, b0, (short)0, c, false, false);
    c = __builtin_amdgcn_wmma_f32_16x16x4_f32(false, a1, false, b1, (short)0, c, false, false);
    c = __builtin_amdgcn_wmma_f32_16x16x4_f32(false, a2, false, b2, (short)0, c, false, false);
    c = __builtin_amdgcn_wmma_f32_16x16x4_f32(false, a3, false, b3, (short)0, c, false, false);

    // D tile is a contiguous 1KB block: out1[o][ni][cin0 + row][no]
    float* po = out1 + ((((size_t)o * NI_DIM + ni) * CIN_DIM + cin0) << 4)
                     + ((size_t)h << 7) + n;   // row = v + 8h, col = n
#pragma unroll
    for (int v = 0; v < 8; ++v)
      __builtin_nontemporal_store(c[v], po + (size_t)v * 16);
  }
}

// ---------------------------------------------------------------------------
// Output 2: rotated bilinear resampling of the 7x7 spatial kernels (1.6 MB).
// One block per (o, no); 49 active lanes.
// ---------------------------------------------------------------------------
__global__ void __launch_bounds__(64)
gsep_rot_bilinear_kernel(const float* __restrict__ weight, // [O,1,KS,KS]
                         const float* __restrict__ outH,   // [NO]
                         const float* __restrict__ gridRn, // [KS,KS,2]
                         const float* __restrict__ mask,   // [KS,KS]
                         float* __restrict__ out2)         // [O,NO,1,KS,KS]
{
  const int b  = blockIdx.x;            // o*NO + no
  const int no = b & (NO_DIM - 1);
  const int o  = b >> 4;
  const int t  = threadIdx.x;
  if (t >= KS_DIM * KS_DIM) return;

  const float ang = -outH[no];
  const float cs = cosf(ang), sn = sinf(ang);
  const float gx = gridRn[2 * t + 0];
  const float gy = gridRn[2 * t + 1];
  // R = [[c,-s],[s,c]] applied to (gx,gy)
  const float x = cs * gx - sn * gy;
  const float y = sn * gx + cs * gy;

  const float xf  = (x + 1.0f) * 0.5f * (float)(KS_DIM - 1);
  const float yf  = (y + 1.0f) * 0.5f * (float)(KS_DIM - 1);
  const float x0f = floorf(xf), y0f = floorf(yf);
  const float wx  = xf - x0f,   wy  = yf - y0f;
  const int   x0  = (int)x0f,   y0  = (int)y0f;

  const float* img = weight + (size_t)o * (KS_DIM * KS_DIM);

  auto gat = [&](int yy, int xx) -> float {
    int yc = min(max(yy, 0), KS_DIM - 1);
    int xc = min(max(xx, 0), KS_DIM - 1);
    float v = img[yc * KS_DIM + xc];
    bool ok = (yy >= 0) && (yy < KS_DIM) && (xx >= 0) && (xx < KS_DIM);
    return ok ? v : 0.0f;
  };

  float val = (1.0f - wy) * (1.0f - wx) * gat(y0,     x0    )
            + (1.0f - wy) * wx          * gat(y0,     x0 + 1)
            + wy          * (1.0f - wx) * gat(y0 + 1, x0    )
            + wy          * wx          * gat(y0 + 1, x0 + 1);

  __builtin_nontemporal_store(val * mask[t],
                              out2 + (size_t)b * (KS_DIM * KS_DIM) + t);
}

extern "C" void kernel_launch(void* const* d_in, const int* in_sizes, int n_in,
                              void* d_out, int out_size, void* d_ws, size_t ws_size,
                              hipStream_t stream) {
  const float* in_H     = (const float*)d_in[0];  // [NI,1]
  const float* out_H    = (const float*)d_in[1];  // [NO,1]
  const float* weight_H = (const float*)d_in[2];  // [O,CIN,K]
  const float* weight   = (const float*)d_in[3];  // [O,1,KS,KS]
  // d_in[4] = grid_H (implicit uniform grid; not needed)
  const float* grid_Rn  = (const float*)d_in[5];  // [KS,KS,2]
  const float* mask     = (const float*)d_in[6];  // [KS,KS]

  float* out1 = (float*)d_out;                                    // [O,NI,CIN,NO]
  float* out2 = out1 + (size_t)O_DIM * NI_DIM * CIN_DIM * NO_DIM; // [O,NO,1,KS,KS]

  // (O*CIN)/16 = 16384 row tiles, 8 waves per 256-thread block -> 2048 blocks
  gsep_lerp_wmma_kernel<<<(O_DIM * CIN_DIM / 16) / 8, 256, 0, stream>>>(
      weight_H, in_H, out_H, out1);

  gsep_rot_bilinear_kernel<<<O_DIM * NO_DIM, 64, 0, stream>>>(
      weight, out_H, grid_Rn, mask, out2);
}